// Hybrid_block_64175401337035
// MI455X (gfx1250) — compile-verified
//
#include <hip/hip_runtime.h>
#include <stdint.h>

typedef __attribute__((ext_vector_type(2))) float v2f;
typedef __attribute__((ext_vector_type(8))) float v8f;
typedef __attribute__((ext_vector_type(4))) unsigned int v4u;
typedef __attribute__((ext_vector_type(8))) int v8i;
typedef __attribute__((ext_vector_type(4))) int v4i;

namespace {
constexpr int LEN        = 32768;   // time length
constexpr int NF         = 17;      // filters
constexpr int MAXK       = 96;      // padded kernel length
constexpr int PAD_LO     = 47;      // (MAXK-1)/2
constexpr int TPB        = 256;     // 8 waves (wave32)
constexpr int WAVES      = TPB / 32;
constexpr int TILES_PW   = 4;                       // 16-t tiles per wave
constexpr int T_PER_WAVE = TILES_PW * 16;           // 64
constexpr int T_PER_BLK  = WAVES * T_PER_WAVE;      // 512
constexpr int XS_LEN     = T_PER_BLK + MAXK;        // 608 (607 needed)
constexpr int KSTEPS     = MAXK / 4;                // 24 WMMA k-steps
}

#if __has_builtin(__builtin_amdgcn_tensor_load_to_lds) && \
    __has_builtin(__builtin_amdgcn_s_wait_tensorcnt)
#define USE_TDM 1
#else
#define USE_TDM 0
#endif

// C[m=filter 16][n=time 16] = sum_k A[m][k] * B[k][n]
//   A[f][k] = W[f][k]            (hoisted to VGPRs, time-invariant)
//   B[k][n] = xs[tile + n + k]   (implicit im2col window in LDS)
__global__ __launch_bounds__(TPB) void conv17_wmma_f32(
    const float* __restrict__ x, const float* __restrict__ W,
    float* __restrict__ out) {
  __shared__ float xs[XS_LEN];

  const int b   = blockIdx.y;
  const int t0  = blockIdx.x * T_PER_BLK;
  const int tid = threadIdx.x;
  const int lane = tid & 31;
  const int wave = tid >> 5;
  const int hi   = lane >> 4;   // half-wave select (K offset 0/2, M offset 0/8)
  const int ln   = lane & 15;

  const float* xb = x + (size_t)b * LEN;

  // ---- Stage x window [t0-47, t0+512+48) into LDS -------------------------
  const int start  = t0 - PAD_LO;              // may be negative (block x==0)
  const int lo     = start < 0 ? -start : 0;   // front elements to zero-fill
  const int gstart = start + lo;               // first in-range element

  // Zero left pad (only block x==0 does any work here).
  for (int i = tid; i < lo; i += TPB) xs[i] = 0.0f;

#if USE_TDM
  if (wave == 0) {
    // Tensor DMA Descriptor (D#), per CDNA5 ISA ch.8.
    const uint64_t ga = (uint64_t)(uintptr_t)(const void*)(xb + gstart);
    const uint32_t lds_addr = (uint32_t)(uintptr_t)(void*)&xs[0] +
                              (uint32_t)lo * 4u;
    const uint32_t td0   = (uint32_t)(LEN - gstart);   // valid elems; OOB -> 0
    const uint32_t tile0 = (uint32_t)(XS_LEN - lo);    // elems to transfer

    v4u g0;
    g0.x = 1u;                                          // count=1 (valid D#)
    g0.y = lds_addr;                                    // lds_addr [63:32]
    g0.z = (uint32_t)(ga & 0xffffffffu);                // global_addr [95:64]
    g0.w = (uint32_t)((ga >> 32) & 0x01ffffffu)         // global_addr [120:96]
         | (2u << 30);                                  // type=2 ("image")

    v8i g1;
    g1[0] = (int)(2u << 16);                            // data_size=2 (4 bytes)
    g1[1] = (int)((td0 & 0xffffu) << 16);               // tensor_dim0[15:0]
    g1[2] = (int)(((td0 >> 16) & 0xffffu) | (1u << 16));// td0[31:16] | tdim1=1
    g1[3] = (int)(tile0 << 16);                         // tdim1 hi=0 | tile_dim0
    g1[4] = 1;                                          // tile_dim1=1, tile_dim2=0
    g1[5] = (int)td0;                                   // tensor_dim0_stride lo
    g1[6] = 0;                                          // stride hi | dim1_stride lo
    g1[7] = 0;                                          // dim1_stride hi

    v4i g2 = {0, 0, 0, 0};                              // >2D dims unused
    v4i g3 = {0, 0, 0, 0};

#if defined(__clang_major__) && (__clang_major__ >= 23)
    v8i g4 = {0, 0, 0, 0, 0, 0, 0, 0};
    __builtin_amdgcn_tensor_load_to_lds(g0, g1, g2, g3, g4, 0);
#else
    __builtin_amdgcn_tensor_load_to_lds(g0, g1, g2, g3, 0);
#endif
    __builtin_amdgcn_s_wait_tensorcnt(0);               // DMA -> LDS complete
  }
#else
  for (int i = tid; i < XS_LEN; i += TPB) {
    int gt = t0 + i - PAD_LO;
    xs[i] = (gt >= 0 && gt < LEN) ? xb[gt] : 0.0f;
  }
#endif

  // ---- Hoist A (weights) into registers while the DMA is in flight --------
  // Group 0: f = ln (0..15). Group 1: f = 16+ln, only ln==0 is a real filter.
  v2f A0[KSTEPS], A1[KSTEPS];
#pragma unroll
  for (int s = 0; s < KSTEPS; ++s) {
    const int k = 4 * s + 2 * hi;                 // even -> 8B aligned loads
    A0[s] = *(const v2f*)(W + (size_t)ln * MAXK + k);
    v2f a1 = {0.0f, 0.0f};
    if (ln == 0) a1 = *(const v2f*)(W + (size_t)16 * MAXK + k);
    A1[s] = a1;
  }

  __syncthreads();

  // ---- WMMA core ----------------------------------------------------------
  float* ob = out + (size_t)b * NF * LEN;
  const int wt0 = wave * T_PER_WAVE;              // wave offset inside block

  for (int tt = 0; tt < TILES_PW; ++tt) {
    // B element per lane/component: xs[wt0 + tt*16 + n + (4s + 2*hi + j)]
    const int xbase = wt0 + tt * 16 + ln + 2 * hi;
    v8f c0 = {};
    v8f c1 = {};
#pragma unroll
    for (int s = 0; s < KSTEPS; ++s) {
      v2f bmat;
      bmat.x = xs[xbase + 4 * s];
      bmat.y = xs[xbase + 4 * s + 1];
      c0 = __builtin_amdgcn_wmma_f32_16x16x4_f32(
          false, A0[s], false, bmat, (short)0, c0, false, false);
      c1 = __builtin_amdgcn_wmma_f32_16x16x4_f32(
          false, A1[s], false, bmat, (short)0, c1, false, false);
    }

    // Store with ReLU. N=time sits in lanes -> contiguous 64B runs per row.
    const int tbase = t0 + wt0 + tt * 16;         // global t of tile origin
#pragma unroll
    for (int r = 0; r < 8; ++r) {
      const int f = r + 8 * hi;                   // group-0 filter row
      float v = c0[r];
      v = v > 0.0f ? v : 0.0f;
      ob[(size_t)f * LEN + tbase + ln] = v;
    }
    if (hi == 0) {                                // group-1: only f==16 (M=0 row)
      float v = c1[0];
      v = v > 0.0f ? v : 0.0f;
      ob[(size_t)16 * LEN + tbase + ln] = v;
    }
  }
}

extern "C" void kernel_launch(void* const* d_in, const int* in_sizes, int n_in,
                              void* d_out, int out_size, void* d_ws, size_t ws_size,
                              hipStream_t stream) {
  (void)in_sizes; (void)n_in; (void)d_ws; (void)ws_size; (void)out_size;
  const float* x = (const float*)d_in[0];   // (128, 1, 32768) f32
  const float* W = (const float*)d_in[1];   // (17, 1, 96) f32
  float* out = (float*)d_out;               // (128, 17, 32768) f32

  dim3 grid(LEN / T_PER_BLK, 128, 1);       // (64, 128)
  dim3 block(TPB, 1, 1);
  conv17_wmma_f32<<<grid, block, 0, stream>>>(x, W, out);
}